// DILATEDIVloss_35476429865782
// MI455X (gfx1250) — compile-verified
//
#include <hip/hip_runtime.h>

#define BN 64
#define NN 256
#define TILE 16
#define INF_F 1e10f
#define GAMMA_F 0.01f
#define INV_G 100.0f            // 1/gamma
#define ALPHA_F 0.5f

typedef float v2f __attribute__((ext_vector_type(2)));
typedef float v8f __attribute__((ext_vector_type(8)));

// ---------------------------------------------------------------------------
// Kernel 1: D_xy[b,i,j] = (target[b,i] - input[b,j])^2 via rank-1 WMMA.
// One wave per 16x16 tile: D = A(16x4) x B(4x16) + C with
//   A[m,0] = -2*a_m (K=0 real, K=1..3 zero),  B[0,n] = b_n,  C[m,n] = a_m^2 + b_n^2
// A layout (32b A 16x4): lanes 0-15 hold {K0,K1}, lanes 16-31 hold {K2,K3}.
// B layout (row-striped): VGPR0 lanes 0-15 = K0 row, lanes 16-31 = K2 row (zeros).
// C/D layout: VGPR v holds M = v + 8*(lane>>4), N = lane&15.
// ---------------------------------------------------------------------------
__global__ __launch_bounds__(256) void dilate_cost_wmma(
    const float* __restrict__ input, const float* __restrict__ target,
    float* __restrict__ D)
{
  int wave = (blockIdx.x << 3) + (threadIdx.x >> 5);   // 8 waves / block
  int lane = threadIdx.x & 31;
  int b  = wave >> 8;                                  // 16*16 = 256 tiles / batch
  int t  = wave & 255;
  int tr = t >> 4;                                     // tile row (target rows)
  int tc = t & 15;                                     // tile col (input cols)

  const float* arow = target + b * NN + tr * TILE;     // 16 row features
  const float* bcol = input  + b * NN + tc * TILE;     // 16 col features

  int hi = lane >> 4;
  int lo = lane & 15;

  v2f A;                                               // A-matrix 16x4, only K=0 live
  A.x = hi ? 0.0f : -2.0f * arow[lo];
  A.y = 0.0f;
  v2f Bm;                                              // B-matrix 4x16, only K=0 live
  Bm.x = hi ? 0.0f : bcol[lo];
  Bm.y = 0.0f;

  float bn  = bcol[lo];
  float bn2 = bn * bn;
  v8f C;
  #pragma unroll
  for (int v = 0; v < 8; ++v) {
    float am = arow[v + 8 * hi];
    C[v] = am * am + bn2;
  }

  // (neg_a, A, neg_b, B, c_mod, C, reuse_a, reuse_b)
  C = __builtin_amdgcn_wmma_f32_16x16x4_f32(false, A, false, Bm, (short)0, C,
                                            false, false);

  float* out = D + ((size_t)b * NN + (size_t)tr * TILE) * NN + tc * TILE + lo;
  #pragma unroll
  for (int v = 0; v < 8; ++v) {
    int m = v + 8 * hi;
    out[(size_t)m * NN] = fmaxf(C[v], 0.0f);
  }
}

// ---------------------------------------------------------------------------
// Kernel 2: anti-diagonal soft-DTW forward. 192 blocks = (which 0..2) x batch.
// which==0 : D_xy streamed from L2 (precomputed tile kernel), R written out.
// which==1 : D_xx on the fly (target vs target).  which==2 : D_yy (input).
// 3-deep LDS ring of diagonals, one barrier per wavefront step.
// ---------------------------------------------------------------------------
__global__ __launch_bounds__(256) void dilate_sdtw_fwd(
    const float* __restrict__ input, const float* __restrict__ target,
    const float* __restrict__ D, float* __restrict__ R,
    float* __restrict__ sdtw)
{
  __shared__ float sb[3][NN];
  __shared__ float bsh[NN];
  int i     = threadIdx.x;
  int which = blockIdx.x >> 6;          // 0=xy 1=xx 2=yy
  int b     = blockIdx.x & 63;

  float ai = (which == 2) ? input[b * NN + i] : target[b * NN + i];
  bsh[i]   = (which == 1) ? target[b * NN + i] : input[b * NN + i];
  sb[0][i] = INF_F; sb[1][i] = INF_F; sb[2][i] = INF_F;
  __syncthreads();

  const float* Drow = D + ((size_t)b * NN + i) * NN;
  float*       Rrow = R + ((size_t)b * NN + i) * NN;

  float last = INF_F;
  #pragma unroll 1
  for (int k = 0; k < 2 * NN - 1; ++k) {
    float* r2 = sb[k % 3];
    float* r1 = sb[(k + 1) % 3];
    float* rn = sb[(k + 2) % 3];
    int  jj    = k - i;
    bool valid = (jj >= 0) && (jj < NN);

    float d;
    if (which == 0) {
      d = valid ? Drow[jj] : 0.0f;
    } else {
      float diff = ai - (valid ? bsh[jj] : 0.0f);
      d = valid ? diff * diff : 0.0f;
    }

    float r2m  = (i > 0) ? r2[i - 1] : INF_F;       // R[i-1,j-1]
    float up   = (i > 0) ? r1[i - 1] : INF_F;       // R[i-1,j]
    float left = (jj == 0) ? INF_F : r1[i];         // R[i,j-1]
    float diag = (i == 0 && k == 0) ? 0.0f : r2m;   // R[-1,-1] = 0 start

    float m = fminf(diag, fminf(up, left));
    float s = __expf((m - diag) * INV_G) + __expf((m - up) * INV_G) +
              __expf((m - left) * INV_G);
    float rnew = d + m - GAMMA_F * __logf(s);
    if (!valid) rnew = INF_F;

    rn[i] = rnew;
    if (valid) {
      last = rnew;
      if (which == 0) Rrow[jj] = rnew;
    }
    __syncthreads();
  }
  if (i == NN - 1) sdtw[which * BN + b] = last;      // R[N-1,N-1]
}

// ---------------------------------------------------------------------------
// Kernel 3: soft-DTW backward on D_xy, fused with the E*Omega contraction.
// E[i,j] = sum over successors of E_succ * exp((R_succ - R[i,j] - D_succ)/g).
// All exponents <= 0 (softmin inequality) so __expf cannot overflow.
// ---------------------------------------------------------------------------
__global__ __launch_bounds__(256) void dilate_sdtw_bwd(
    const float* __restrict__ D, const float* __restrict__ R,
    float* __restrict__ partial)
{
  __shared__ float eb[3][NN];
  __shared__ float red[NN];
  int i = threadIdx.x;
  int b = blockIdx.x;
  eb[0][i] = 0.0f; eb[1][i] = 0.0f; eb[2][i] = 0.0f;
  __syncthreads();

  const float* Rb = R + (size_t)b * NN * NN;
  const float* Db = D + (size_t)b * NN * NN;

  float tacc = 0.0f;
  #pragma unroll 1
  for (int k = 2 * NN - 2; k >= 0; --k) {
    float* en = eb[k % 3];
    float* e1 = eb[(k + 1) % 3];       // diagonal k+1
    float* e2 = eb[(k + 2) % 3];       // diagonal k+2
    int  jj    = k - i;
    bool valid = (jj >= 0) && (jj < NN);

    float e = 0.0f;
    if (k == 2 * NN - 2) {
      e = (i == NN - 1) ? 1.0f : 0.0f; // seed E[N-1,N-1] = 1
    } else if (valid) {
      float rij = Rb[(size_t)i * NN + jj];
      if (i + 1 < NN) {                                    // E[i+1,j]
        float w = __expf((Rb[(size_t)(i + 1) * NN + jj] - rij -
                          Db[(size_t)(i + 1) * NN + jj]) * INV_G);
        e += e1[i + 1] * w;
      }
      if (jj + 1 < NN) {                                   // E[i,j+1]
        float w = __expf((Rb[(size_t)i * NN + jj + 1] - rij -
                          Db[(size_t)i * NN + jj + 1]) * INV_G);
        e += e1[i] * w;
      }
      if (i + 1 < NN && jj + 1 < NN) {                     // E[i+1,j+1]
        float w = __expf((Rb[(size_t)(i + 1) * NN + jj + 1] - rij -
                          Db[(size_t)(i + 1) * NN + jj + 1]) * INV_G);
        e += e2[i + 1] * w;
      }
    }
    en[i] = valid ? e : 0.0f;
    if (valid) {
      int dd = i - jj;                 // Omega = (i-j)^2 (t-offset cancels)
      tacc += e * (float)(dd * dd);
    }
    __syncthreads();
  }

  red[i] = tacc;
  __syncthreads();
  for (int s = NN / 2; s > 0; s >>= 1) {
    if (i < s) red[i] += red[i + s];
    __syncthreads();
  }
  if (i == 0) partial[b] = red[0];
}

// ---------------------------------------------------------------------------
// Kernel 4: deterministic final reduction (no atomics -> graph-replay safe).
// ---------------------------------------------------------------------------
__global__ __launch_bounds__(64) void dilate_final(
    const float* __restrict__ sdtw, const float* __restrict__ partial,
    float* __restrict__ out)
{
  __shared__ float s1[64], s2[64];
  int i = threadIdx.x;
  s1[i] = sdtw[i] - 0.5f * (sdtw[BN + i] + sdtw[2 * BN + i]);
  s2[i] = partial[i];
  __syncthreads();
  for (int s = 32; s > 0; s >>= 1) {
    if (i < s) { s1[i] += s1[i + s]; s2[i] += s2[i + s]; }
    __syncthreads();
  }
  if (i == 0) {
    float loss_shape    = s1[0] / (float)BN;
    float loss_temporal = (s2[0] / (float)BN) / (float)(NN * NN);
    out[0] = ALPHA_F * loss_shape + (1.0f - ALPHA_F) * loss_temporal;
  }
}

extern "C" void kernel_launch(void* const* d_in, const int* in_sizes, int n_in,
                              void* d_out, int out_size, void* d_ws, size_t ws_size,
                              hipStream_t stream) {
  const float* input  = (const float*)d_in[0];   // [B, N, 1]
  const float* target = (const float*)d_in[1];   // [B, N, 1]

  char*  ws      = (char*)d_ws;
  size_t matB    = (size_t)BN * NN * NN * sizeof(float);   // 16.78 MB
  float* D       = (float*)ws;                             // D_xy
  float* R       = (float*)(ws + matB);                    // soft-DTW R for xy
  float* sdtw    = (float*)(ws + 2 * matB);                // [3][64]
  float* partial = sdtw + 3 * BN;                          // [64]
  float* out     = (float*)d_out;

  // 64 batches * 256 tiles = 16384 waves, 8 waves per 256-thread block
  dilate_cost_wmma<<<2048, 256, 0, stream>>>(input, target, D);
  // 3 matrices * 64 batches, one workgroup-wavefront DP each
  dilate_sdtw_fwd<<<192, 256, 0, stream>>>(input, target, D, R, sdtw);
  // backward + E*Omega contraction, one block per batch
  dilate_sdtw_bwd<<<64, 256, 0, stream>>>(D, R, partial);
  // final scalar
  dilate_final<<<1, 64, 0, stream>>>(sdtw, partial, out);
}